// CompLayer_7550552506735
// MI455X (gfx1250) — compile-verified
//
#include <hip/hip_runtime.h>
#include <hip/hip_bf16.h>
#include <math.h>

#define N_NODES 50000
#define N_EDGES 1250000
#define EMB     64
#define BN_EPS  1e-5f

typedef __attribute__((ext_vector_type(2))) float v2f;
typedef __attribute__((ext_vector_type(8))) float v8f;

// ---- ordered-uint encoding for float atomic max (handles negatives) ----
__device__ __forceinline__ unsigned f2ord(float f) {
    unsigned u = __float_as_uint(f);
    return (u & 0x80000000u) ? ~u : (u | 0x80000000u);
}
__device__ __forceinline__ float ord2f(unsigned u) {
    return (u & 0x80000000u) ? __uint_as_float(u & 0x7fffffffu)
                             : __uint_as_float(~u);
}

// ---- init scratch: smax_ord = 0 (== -NaN sentinel, below any real ord), ssum=0, neigh=0
__global__ void k_init(unsigned* __restrict__ smax_ord,
                       float* __restrict__ ssum,
                       float* __restrict__ neigh) {
    int i = blockIdx.x * blockDim.x + threadIdx.x;
    if (i < N_NODES) { smax_ord[i] = 0u; ssum[i] = 0.0f; }
    if (i < N_NODES * EMB) neigh[i] = 0.0f;
}

// ---- per-edge score + segment max (one wave32 per edge) ----
__global__ void k_score(const float* __restrict__ ent, const float* __restrict__ rel,
                        const int* __restrict__ src, const int* __restrict__ dst,
                        const int* __restrict__ rid,
                        float* __restrict__ score, unsigned* __restrict__ smax_ord) {
    int e    = (blockIdx.x * blockDim.x + threadIdx.x) >> 5;
    int lane = threadIdx.x & 31;
    if (e >= N_EDGES) return;
    int s = src[e], d = dst[e], r = rid[e];
    const float* es = ent + (size_t)s * EMB;
    const float* ed = ent + (size_t)d * EMB;
    const float* rr = rel + (size_t)r * EMB;
    float p = es[lane]      * rr[lane]      * ed[lane]
            + es[lane + 32] * rr[lane + 32] * ed[lane + 32];
#pragma unroll
    for (int off = 16; off > 0; off >>= 1) p += __shfl_xor(p, off, 32);
    if (lane == 0) {
        score[e] = p;
        atomicMax(&smax_ord[d], f2ord(p));
    }
}

// ---- e = exp(score - smax[dst]); segment sum ----
__global__ void k_exp(const float* __restrict__ score, const int* __restrict__ dst,
                      const unsigned* __restrict__ smax_ord,
                      float* __restrict__ expe, float* __restrict__ ssum) {
    int i = blockIdx.x * blockDim.x + threadIdx.x;
    if (i >= N_EDGES) return;
    int d = dst[i];
    float ev = expf(score[i] - ord2f(smax_ord[d]));
    expe[i] = ev;
    atomicAdd(&ssum[d], ev);
}

// ---- weighted scatter: neigh[dst] += comp * alpha (64 lanes per edge) ----
__global__ void k_scatter(const float* __restrict__ ent, const float* __restrict__ rel,
                          const int* __restrict__ src, const int* __restrict__ dst,
                          const int* __restrict__ rid,
                          const float* __restrict__ expe, const float* __restrict__ ssum,
                          float* __restrict__ neigh) {
    int e = blockIdx.x * 4 + (threadIdx.x >> 6);
    int c = threadIdx.x & 63;
    if (e >= N_EDGES) return;
    int s = src[e], d = dst[e], r = rid[e];
    float alpha = expe[e] / ssum[d];
    float comp  = ent[(size_t)s * EMB + c] * rel[(size_t)r * EMB + c];
    atomicAdd(&neigh[(size_t)d * EMB + c], comp * alpha);
}

// ---- h = neigh @ W via V_WMMA_F32_16X16X4_F32 (exact f32 path) ----
// One wave per 16x16 output tile; block = 128 threads = 4 waves = 4 n-tiles.
__global__ void k_gemm_wmma(const float* __restrict__ neigh,
                            const float* __restrict__ W,
                            float* __restrict__ h) {
    int row0  = blockIdx.x * 16;          // 3125 row tiles (50000 = 3125*16 exact)
    int wave  = threadIdx.x >> 5;         // n-tile 0..3
    int lane  = threadIdx.x & 31;
    int n0    = wave * 16;
    int m     = lane & 15;                // A row within tile
    int khalf = lane >> 4;                // 0: K=0,1 ; 1: K=2,3 (ISA 16x4 f32 layout)
    int n     = lane & 15;                // B/D column within tile

    v8f acc = {};
#pragma unroll
    for (int k = 0; k < EMB; k += 4) {
        int ka = k + khalf * 2;
        v2f a, b;
        a.x = neigh[(size_t)(row0 + m) * EMB + ka];
        a.y = neigh[(size_t)(row0 + m) * EMB + ka + 1];
        b.x = W[(size_t)ka       * EMB + n0 + n];
        b.y = W[(size_t)(ka + 1) * EMB + n0 + n];
        acc = __builtin_amdgcn_wmma_f32_16x16x4_f32(
                  false, a, false, b, (short)0, acc, false, false);
    }
    // D layout: lanes 0-15 hold M=0..7 in v0..v7; lanes 16-31 hold M=8..15
    int mbase = row0 + khalf * 8;
#pragma unroll
    for (int r = 0; r < 8; ++r)
        h[(size_t)(mbase + r) * EMB + n0 + n] = acc[r];
}

// ---- per-column BN stats folded into scale/shift ----
__global__ void k_stats(const float* __restrict__ h,
                        const float* __restrict__ gamma, const float* __restrict__ beta,
                        float* __restrict__ scale, float* __restrict__ shift) {
    int c = blockIdx.x;                    // 0..63
    __shared__ float s_sum[256], s_sq[256];
    float sum = 0.0f, sq = 0.0f;
    for (int r = threadIdx.x; r < N_NODES; r += blockDim.x) {
        float v = h[(size_t)r * EMB + c];
        sum += v; sq += v * v;
    }
    s_sum[threadIdx.x] = sum; s_sq[threadIdx.x] = sq;
    __syncthreads();
    for (int off = 128; off > 0; off >>= 1) {
        if (threadIdx.x < off) {
            s_sum[threadIdx.x] += s_sum[threadIdx.x + off];
            s_sq [threadIdx.x] += s_sq [threadIdx.x + off];
        }
        __syncthreads();
    }
    if (threadIdx.x == 0) {
        float mu   = s_sum[0] / (float)N_NODES;
        float var  = s_sq[0] / (float)N_NODES - mu * mu;   // biased var, as torch BN
        float rinv = rsqrtf(var + BN_EPS);
        float sc   = gamma[c] * rinv;
        scale[c] = sc;
        shift[c] = beta[c] - mu * sc;
    }
}

// ---- out = tanh(h*scale + shift), in place on d_out ----
__global__ void k_apply(float* __restrict__ h,
                        const float* __restrict__ scale, const float* __restrict__ shift) {
    int i = blockIdx.x * blockDim.x + threadIdx.x;
    if (i >= N_NODES * EMB) return;
    int c = i & 63;
    h[i] = tanhf(h[i] * scale[c] + shift[c]);
}

extern "C" void kernel_launch(void* const* d_in, const int* in_sizes, int n_in,
                              void* d_out, int out_size, void* d_ws, size_t ws_size,
                              hipStream_t stream) {
    const float* ent   = (const float*)d_in[0];   // [N, 64]
    const float* rel   = (const float*)d_in[1];   // [1000*2? -> N_REL2, 64]
    const int*   src   = (const int*)  d_in[2];   // [E]
    const int*   dst   = (const int*)  d_in[3];   // [E]
    const int*   rid   = (const int*)  d_in[4];   // [E]
    const float* W     = (const float*)d_in[5];   // [64, 64]
    const float* gamma = (const float*)d_in[6];   // [64]
    const float* beta  = (const float*)d_in[7];   // [64]
    float* out = (float*)d_out;                   // [N, 64], also used as h

    // workspace layout (floats)
    float*    ws       = (float*)d_ws;
    float*    score    = ws;                                   // E
    float*    expe     = ws + N_EDGES;                         // E
    unsigned* smax_ord = (unsigned*)(ws + 2 * (size_t)N_EDGES);// N
    float*    ssum     = ws + 2 * (size_t)N_EDGES + N_NODES;   // N
    float*    neigh    = ws + 2 * (size_t)N_EDGES + 2 * (size_t)N_NODES; // N*64
    float*    scale    = neigh + (size_t)N_NODES * EMB;        // 64
    float*    shift    = scale + EMB;                          // 64

    k_init   <<<(N_NODES * EMB + 255) / 256, 256, 0, stream>>>(smax_ord, ssum, neigh);
    k_score  <<<N_EDGES / 8,  256, 0, stream>>>(ent, rel, src, dst, rid, score, smax_ord);
    k_exp    <<<(N_EDGES + 255) / 256, 256, 0, stream>>>(score, dst, smax_ord, expe, ssum);
    k_scatter<<<N_EDGES / 4,  256, 0, stream>>>(ent, rel, src, dst, rid, expe, ssum, neigh);
    k_gemm_wmma<<<N_NODES / 16, 128, 0, stream>>>(neigh, W, out);
    k_stats  <<<EMB, 256, 0, stream>>>(out, gamma, beta, scale, shift);
    k_apply  <<<(N_NODES * EMB + 255) / 256, 256, 0, stream>>>(out, scale, shift);
}